// DecoderZX_15582141350495
// MI455X (gfx1250) — compile-verified
//
#include <hip/hip_runtime.h>
#include <hip/hip_bf16.h>

typedef __attribute__((ext_vector_type(2))) float v2f;
typedef __attribute__((ext_vector_type(8))) float v8f;

constexpr int B_ROWS   = 2048;
constexpr int N_LAT    = 64;
constexpr int N_NUI    = 16;
constexpr int N_OUT    = 2048;
constexpr int Z_STRIDE = N_LAT + N_NUI;   // 80
constexpr int MAX_TILES = 144;            // sum ceil(c_s/16) <= 143
constexpr int PERM_CAP  = 2304;           // padded row slots (<= 2288)

// Swizzled combined-weight buffer: v2f swz[((sid*16 + kc)*2 + h)*N_OUT + col]
//   = { Wc[kc*4+2h][col], Wc[kc*4+2h+1][col] },  Wc = amat_w + amat_site[sid]
constexpr int    SWZ_ELEMS  = N_NUI * 16 * 2 * N_OUT;          // 1,048,576 v2f
constexpr size_t SWZ_OFF    = 16384;                            // bytes into ws
constexpr size_t WS_NEEDED  = SWZ_OFF + (size_t)SWZ_ELEMS * sizeof(v2f); // ~8.4MB

// ---------------------------------------------------------------------------
// Kernel 1: per-row nuisance id (argmax of last 16 z entries), then group rows
// by id into an id-sorted permutation padded to 16-row tiles. Single block.
// mu is bitwise independent of intra-group ordering -> atomic scatter is safe.
// ---------------------------------------------------------------------------
__global__ __launch_bounds__(256)
void group_kernel(const float* __restrict__ z, int* __restrict__ perm,
                  int* __restrict__ tile_sid, int* __restrict__ n_tiles)
{
  __shared__ int ids[B_ROWS];
  __shared__ int cnt[N_NUI];
  __shared__ int base[N_NUI];
  __shared__ int off[N_NUI];
  const int tid = threadIdx.x;
  if (tid < N_NUI) { cnt[tid] = 0; off[tid] = 0; }
  __syncthreads();
  for (int b = tid; b < B_ROWS; b += 256) {
    const float* p = z + (size_t)b * Z_STRIDE + N_LAT;
    float best = p[0]; int id = 0;
    for (int j = 1; j < N_NUI; ++j) { float v = p[j]; if (v > best) { best = v; id = j; } }
    ids[b] = id;
    atomicAdd(&cnt[id], 1);
  }
  __syncthreads();
  if (tid == 0) {
    int running = 0, t = 0;
    for (int s = 0; s < N_NUI; ++s) {
      base[s] = running;
      int nt = (cnt[s] + 15) >> 4;
      for (int k = 0; k < nt; ++k) tile_sid[t++] = s;
      running += nt << 4;
    }
    *n_tiles = t;
  }
  __syncthreads();
  for (int i = tid; i < PERM_CAP; i += 256) perm[i] = -1;
  __syncthreads();
  for (int b = tid; b < B_ROWS; b += 256) {
    int id = ids[b];
    int pos = base[id] + atomicAdd(&off[id], 1);
    perm[pos] = b;
  }
}

// ---------------------------------------------------------------------------
// Kernel 2: theta = exp(px_r), written after the mu block.
// ---------------------------------------------------------------------------
__global__ __launch_bounds__(256)
void theta_kernel(const float* __restrict__ px_r, float* __restrict__ out)
{
  int o = blockIdx.x * 256 + threadIdx.x;
  if (o < N_OUT) out[(size_t)B_ROWS * N_OUT + o] = __expf(px_r[o]);
}

// ---------------------------------------------------------------------------
// Kernel 2b: precompute swizzled combined weights (one b64 per B-fragment
// lane in the GEMM). Coalesced: consecutive threads -> consecutive cols.
// idx = sid*65536 + kc*4096 + h*2048 + col
// ---------------------------------------------------------------------------
__global__ __launch_bounds__(256)
void combine_kernel(const float* __restrict__ amat_w,
                    const float* __restrict__ amat_site,
                    v2f* __restrict__ swz)
{
  int idx = blockIdx.x * 256 + threadIdx.x;
  int col = idx & (N_OUT - 1);
  int h   = (idx >> 11) & 1;
  int kc  = (idx >> 12) & 15;
  int sid = idx >> 16;
  int k   = kc * 4 + 2 * h;
  const float* w = amat_w + (size_t)k * N_OUT + col;
  const float* s = amat_site + ((size_t)sid * N_LAT + k) * N_OUT + col;
  v2f v;
  v.x = w[0]     + s[0];
  v.y = w[N_OUT] + s[N_OUT];
  swz[idx] = v;
}

// ---------------------------------------------------------------------------
// Kernel 3: fused grouped-GEMM (fp32 WMMA 16x16x4) + offsets + row softmax
// + size_factor scale. One workgroup = 16 same-id rows x 2048 cols.
// 8 waves; wave w owns cols [256w, 256w+256): 16 N-tiles x 16 K-chunks.
// USE_SWZ: B fragments come from the precomputed swizzled buffer (1x b64)
// instead of 4x b32 + 2x add.
// ---------------------------------------------------------------------------
template <bool USE_SWZ>
__global__ __launch_bounds__(256)
void decoder_kernel(const float* __restrict__ z, const float* __restrict__ sf,
                    const float* __restrict__ amat_w, const float* __restrict__ amat_site,
                    const v2f* __restrict__ swz, const float* __restrict__ offsets,
                    const int* __restrict__ perm, const int* __restrict__ tile_sid,
                    const int* __restrict__ n_tiles, float* __restrict__ out)
{
  const int mt = blockIdx.x;
  if (mt >= *n_tiles) return;

  const int tid     = threadIdx.x;
  const int wave    = tid >> 5;
  const int lane    = tid & 31;
  const int halfsel = lane >> 4;   // 0: lanes 0-15, 1: lanes 16-31
  const int mrow    = lane & 15;   // M for A-frag / N for B-frag / N for C-frag
  const int sid     = tile_sid[mt];

  __shared__ float zt[16][68];     // padded to dodge bank conflicts
  __shared__ int   rows[16];
  __shared__ float red[16][8];
  __shared__ float rowstat[16];
  __shared__ float rowscale[16];

  if (tid < 16) rows[tid] = perm[mt * 16 + tid];
  __syncthreads();
  for (int i = tid; i < 16 * 64; i += 256) {
    int r = i >> 6, c = i & 63;
    int b = rows[r];
    zt[r][c] = (b >= 0) ? z[(size_t)b * Z_STRIDE + c] : 0.0f;
  }
  __syncthreads();

  // A fragments: 16x4 f32 tile per K-chunk. VGPR0: K=k0 (lanes0-15), K=k0+2
  // (lanes16-31); VGPR1: K=k0+1 / K=k0+3.
  v2f afrag[16];
#pragma unroll
  for (int kc = 0; kc < 16; ++kc) {
    int k0 = kc * 4 + halfsel * 2;
    afrag[kc].x = zt[mrow][k0];
    afrag[kc].y = zt[mrow][k0 + 1];
  }

  const int colbase = wave * 256;
  v8f acc[16];
#pragma unroll
  for (int t = 0; t < 16; ++t)
    acc[t] = (v8f){0.f,0.f,0.f,0.f,0.f,0.f,0.f,0.f};

  // GEMM: K-chunk outer, N-tile inner -> 16 independent accumulation chains.
#pragma unroll
  for (int kc = 0; kc < 16; ++kc) {
    const int k = kc * 4 + halfsel * 2;
    const float* w0 = amat_w + (size_t)k * N_OUT;
    const float* s0 = amat_site + ((size_t)sid * N_LAT + k) * N_OUT;
    const v2f* bb = swz + ((size_t)((sid * 16 + kc) * 2 + halfsel)) * N_OUT;
#pragma unroll
    for (int t = 0; t < 16; ++t) {
      int col = colbase + t * 16 + mrow;
      v2f bfrag;
      if constexpr (USE_SWZ) {
        bfrag = bb[col];                       // one global_load_b64
      } else {
        bfrag.x = w0[col]         + s0[col];
        bfrag.y = w0[N_OUT + col] + s0[N_OUT + col];
      }
      acc[t] = __builtin_amdgcn_wmma_f32_16x16x4_f32(
          false, afrag[kc], false, bfrag, (short)0, acc[t], false, false);
    }
  }

  // + offsets[sid, col] (broadcast down the 8 rows held per VGPR)
#pragma unroll
  for (int t = 0; t < 16; ++t) {
    float off = offsets[(size_t)sid * N_OUT + colbase + t * 16 + mrow];
#pragma unroll
    for (int i = 0; i < 8; ++i) acc[t][i] += off;
  }

  // ---- row max: per-lane over tiles, butterfly over 16-lane half, LDS over waves
  float pm[8];
#pragma unroll
  for (int r = 0; r < 8; ++r) {
    float m = acc[0][r];
#pragma unroll
    for (int t = 1; t < 16; ++t) m = fmaxf(m, acc[t][r]);
    for (int mask = 1; mask <= 8; mask <<= 1) m = fmaxf(m, __shfl_xor(m, mask, 32));
    pm[r] = m;
  }
  if (mrow == 0) {
#pragma unroll
    for (int r = 0; r < 8; ++r) red[halfsel * 8 + r][wave] = pm[r];
  }
  __syncthreads();
  if (tid < 16) {
    float m = red[tid][0];
    for (int w = 1; w < 8; ++w) m = fmaxf(m, red[tid][w]);
    rowstat[tid] = m;
  }
  __syncthreads();

  // ---- exp in place + row sum
  float ps[8];
#pragma unroll
  for (int r = 0; r < 8; ++r) {
    float rm = rowstat[halfsel * 8 + r];
    float s = 0.0f;
#pragma unroll
    for (int t = 0; t < 16; ++t) {
      float e = __expf(acc[t][r] - rm);
      acc[t][r] = e;
      s += e;
    }
    for (int mask = 1; mask <= 8; mask <<= 1) s += __shfl_xor(s, mask, 32);
    ps[r] = s;
  }
  if (mrow == 0) {
#pragma unroll
    for (int r = 0; r < 8; ++r) red[halfsel * 8 + r][wave] = ps[r];
  }
  __syncthreads();
  if (tid < 16) {
    float s = 0.0f;
    for (int w = 0; w < 8; ++w) s += red[tid][w];
    int b = rows[tid];
    float sfv = (b >= 0) ? sf[b] : 1.0f;
    rowscale[tid] = sfv / s;
  }
  __syncthreads();

  // ---- scaled store (skip padded dummy rows); 64B contiguous per half-wave
#pragma unroll
  for (int r = 0; r < 8; ++r) {
    int row = halfsel * 8 + r;
    int b = rows[row];
    if (b >= 0) {
      float sc = rowscale[row];
      size_t robase = (size_t)b * N_OUT + colbase + mrow;
#pragma unroll
      for (int t = 0; t < 16; ++t)
        out[robase + t * 16] = acc[t][r] * sc;
    }
  }
}

// ---------------------------------------------------------------------------
extern "C" void kernel_launch(void* const* d_in, const int* in_sizes, int n_in,
                              void* d_out, int out_size, void* d_ws, size_t ws_size,
                              hipStream_t stream) {
  const float* z         = (const float*)d_in[0];
  const float* sf        = (const float*)d_in[1];
  const float* amat_w    = (const float*)d_in[2];
  const float* amat_site = (const float*)d_in[3];
  const float* offsets   = (const float*)d_in[4];
  const float* px_r      = (const float*)d_in[5];
  float* out = (float*)d_out;

  int* perm     = (int*)d_ws;           // PERM_CAP ints
  int* tile_sid = perm + PERM_CAP;      // MAX_TILES ints (+pad)
  int* ntiles   = tile_sid + MAX_TILES; // 1 int
  v2f* swz      = (v2f*)((char*)d_ws + SWZ_OFF);

  group_kernel<<<1, 256, 0, stream>>>(z, perm, tile_sid, ntiles);
  theta_kernel<<<(N_OUT + 255) / 256, 256, 0, stream>>>(px_r, out);

  if (ws_size >= WS_NEEDED) {
    combine_kernel<<<SWZ_ELEMS / 256, 256, 0, stream>>>(amat_w, amat_site, swz);
    decoder_kernel<true><<<MAX_TILES, 256, 0, stream>>>(
        z, sf, amat_w, amat_site, swz, offsets, perm, tile_sid, ntiles, out);
  } else {
    decoder_kernel<false><<<MAX_TILES, 256, 0, stream>>>(
        z, sf, amat_w, amat_site, swz, offsets, perm, tile_sid, ntiles, out);
  }
}